// noCluster_Mean_n_Max_16183436771705
// MI455X (gfx1250) — compile-verified
//
#include <hip/hip_runtime.h>
#include <hip/hip_bf16.h>
#include <math.h>

// Shapes from the reference:
//   feature_seq: [T]   int32
//   offset_seq : [B]   int32 (sorted bag starts, offset[0]==0)
//   W          : [VOCAB, 128]  fp32
//   L          : [128, 256]    fp32  (TYPES x 2D)
//   out        : [B, 128]      fp32  = concat(mean, max) @ L.T
#define D 128
#define TWO_D 256
#define NTYPES 128

typedef __attribute__((ext_vector_type(2))) float v2f;
typedef __attribute__((ext_vector_type(8))) float v8f;

// ---------------------------------------------------------------------------
// Kernel 1: per-bag mean+max reduction of gathered embedding rows.
// One 128-thread block per bag; thread tid owns embedding column tid.
// Token ids staged through LDS in chunks of 128 so the id fetch is one
// coalesced load per chunk instead of a redundant uniform load per token.
// ---------------------------------------------------------------------------
__global__ void __launch_bounds__(128)
bag_mean_max_kernel(const int* __restrict__ feat,
                    const int* __restrict__ offs,
                    const float* __restrict__ W,
                    float* __restrict__ men,   // [B, 256] = [mean | max]
                    int T, int B) {
    __shared__ int s_tok[128];
    const int b   = blockIdx.x;
    const int tid = threadIdx.x;

    const int t0 = offs[b];
    const int t1 = (b + 1 < B) ? offs[b + 1] : T;
    const int cnt = t1 - t0;

    float sum = 0.0f;
    float mx  = -INFINITY;

    for (int base = t0; base < t1; base += 128) {
        const int nchunk = min(128, t1 - base);
        __syncthreads();
        if (tid < nchunk) s_tok[tid] = feat[base + tid];
        __syncthreads();
        for (int i = 0; i < nchunk; ++i) {
            const int tok = s_tok[i];
            // prefetch a few rows ahead (emits global_prefetch_b8); table is
            // L2-resident (51 MB < 192 MB L2) so this hides L2 latency.
            if (i + 4 < nchunk) {
                __builtin_prefetch(&W[(size_t)s_tok[i + 4] * D + tid], 0, 1);
            }
            const float v = W[(size_t)tok * D + tid];
            sum += v;
            mx = fmaxf(mx, v);
        }
    }

    const float denom = (float)max(cnt, 1);
    const float meanv = sum / denom;
    const float maxv  = (cnt > 0) ? mx : 0.0f;

    men[(size_t)b * TWO_D + tid]     = meanv;
    men[(size_t)b * TWO_D + D + tid] = maxv;
}

// ---------------------------------------------------------------------------
// Kernel 2: out[B,128] = men[B,256] @ L^T  using V_WMMA_F32_16X16X4_F32.
// Block = 256 threads = 8 waves; block m-tile index = blockIdx.x (16 rows),
// wave w handles n-tile w (16 cols). K=256 swept in steps of 4.
//
// ISA f32 16x4 A-layout (per lane l): row M = l&15,
//   v0 holds K = kbase + (l<16 ? 0 : 2), v1 holds K = kbase + (l<16 ? 1 : 3).
// B (4x16, K x N) mirrors this with N = l&15. C/D 16x16 f32: VGPR r, lane l:
//   l<16 -> (M=r,     N=l), l>=16 -> (M=r+8, N=l-16).
// ---------------------------------------------------------------------------
__global__ void __launch_bounds__(256)
men_linear_wmma_kernel(const float* __restrict__ men,  // [B, 256]
                       const float* __restrict__ L,    // [128, 256]
                       float* __restrict__ out) {      // [B, 128]
    const int lane   = threadIdx.x & 31;
    const int wave   = threadIdx.x >> 5;     // 0..7 -> n tile
    const int m_base = blockIdx.x * 16;
    const int n_base = wave * 16;

    const int r     = lane & 15;             // A row / B col within the tile
    const int khalf = (lane >> 4) * 2;       // 0 or 2

    // 8-byte aligned: row strides are 256 floats, khalf is 0 or 2 floats.
    const float* arow = men + (size_t)(m_base + r) * TWO_D + khalf;
    const float* brow = L   + (size_t)(n_base + r) * TWO_D + khalf;

    v8f c = {};
#pragma unroll
    for (int k = 0; k < TWO_D; k += 4) {
        const v2f a  = *(const v2f*)(arow + k);   // K = k+khalf, k+khalf+1
        const v2f bm = *(const v2f*)(brow + k);
        c = __builtin_amdgcn_wmma_f32_16x16x4_f32(
                /*neg_a=*/false, a, /*neg_b=*/false, bm,
                /*c_mod=*/(short)0, c, /*reuse_a=*/false, /*reuse_b=*/false);
    }

    const int n        = lane & 15;
    const int mrow_off = (lane >> 4) * 8;
#pragma unroll
    for (int v = 0; v < 8; ++v) {
        out[(size_t)(m_base + mrow_off + v) * NTYPES + n_base + n] = c[v];
    }
}

// ---------------------------------------------------------------------------
extern "C" void kernel_launch(void* const* d_in, const int* in_sizes, int n_in,
                              void* d_out, int out_size, void* d_ws, size_t ws_size,
                              hipStream_t stream) {
    const int*   feat = (const int*)  d_in[0];
    const int*   offs = (const int*)  d_in[1];
    const float* W    = (const float*)d_in[2];
    const float* L    = (const float*)d_in[3];
    float*       out  = (float*)d_out;

    const int T = in_sizes[0];
    const int B = in_sizes[1];

    float* men = (float*)d_ws;  // B * 256 floats = 8 MB scratch

    bag_mean_max_kernel<<<B, 128, 0, stream>>>(feat, offs, W, men, T, B);
    men_linear_wmma_kernel<<<B / 16, 256, 0, stream>>>(men, L, out);
}